// DeformableWindowAttention3D_17703855194344
// MI455X (gfx1250) — compile-verified
//
#include <hip/hip_runtime.h>
#include <math.h>

// Problem constants (from reference)
#define B_  4
#define N_  2048
#define C_  192
#define H_  6
#define D_  32
#define KS  16
#define CH  96      // offset-net hidden
#define PH  48      // pos-mlp hidden
#define C3  576     // 3*C
#define NQ  32768   // N*KS queries per batch
#define KT_ (N_/16) // 128 key tiles per batch

typedef __attribute__((ext_vector_type(2))) float v2f;
typedef __attribute__((ext_vector_type(8))) float v8f;

__device__ __forceinline__ float gelu_exact(float v) {
    return 0.5f * v * (1.0f + erff(v * 0.70710678118654752f));
}

// ---------------------------------------------------------------------------
// Kernel 1: offset MLP -> WMMA-swizzled query A-rows and key B-tiles, plus
// positional bias [B,N,K,H]. One block per (b,n).
//
// A-operand lane layout (16x4 f32 WMMA): lanes0-15 hold K0/K1 of row=lane,
// lanes16-31 hold K2/K3 of row=lane-16. Query row q=(n*16+k) -> tile qt=n,
// row-in-tile=k. Per lane float2:
//   lane<16 : (-2sx, -2sy)   lane>=16 : (-2sz, 1)
// B-operand lane layout (4x16): VGPR0 = rows K0|K2, VGPR1 = rows K1|K3:
//   lane<16 : (kx, ky) of key  lane>=16 : (kz, |k|^2) of key
// ---------------------------------------------------------------------------
__global__ void prep_kernel(const float* __restrict__ coords, const float* __restrict__ x,
                            const float* __restrict__ ow1, const float* __restrict__ ob1,
                            const float* __restrict__ ow2, const float* __restrict__ ob2,
                            const float* __restrict__ pw1, const float* __restrict__ pb1,
                            const float* __restrict__ pw2, const float* __restrict__ pb2,
                            float* __restrict__ aswz_, float* __restrict__ kswz_,
                            float* __restrict__ biasb)
{
    __shared__ float xs[C_];
    __shared__ float h1[CH];
    __shared__ float os[KS * 3];
    __shared__ float ph[KS * PH];

    const int bn = blockIdx.x;      // b*N + n
    const int b  = bn >> 11;
    const int n  = bn & (N_ - 1);
    const int t  = threadIdx.x;     // 0..127

    const float* xr = x + (size_t)bn * C_;
    for (int i = t; i < C_; i += 128) xs[i] = xr[i];
    __syncthreads();

    if (t < CH) {
        float acc = ob1[t];
        #pragma unroll 4
        for (int k = 0; k < C_; ++k) acc = fmaf(xs[k], ow1[k * CH + t], acc);
        h1[t] = gelu_exact(acc);
    }
    __syncthreads();

    if (t < KS * 3) {
        float acc = ob2[t];
        #pragma unroll 4
        for (int j = 0; j < CH; ++j) acc = fmaf(h1[j], ow2[j * (KS * 3) + t], acc);
        os[t] = acc;
    }
    __syncthreads();

    const float cx = coords[bn * 3 + 0];
    const float cy = coords[bn * 3 + 1];
    const float cz = coords[bn * 3 + 2];

    if (t < KS) {
        // sample point for (n, k=t); query tile qt == n, row-in-tile == t
        float sx = cx + os[t * 3 + 0] * 10.0f;
        float sy = cy + os[t * 3 + 1] * 10.0f;
        float sz = cz + os[t * 3 + 2] * 10.0f;
        float2* a2 = (float2*)aswz_;
        const size_t base = ((size_t)b * N_ + n) * 32;
        a2[base + t]      = make_float2(-2.0f * sx, -2.0f * sy);
        a2[base + 16 + t] = make_float2(-2.0f * sz, 1.0f);
    }
    if (t == KS) {
        // key tile swizzle: key n -> tile kt=n>>4, slot l=n&15
        float2* k2 = (float2*)kswz_;
        const int  kt = n >> 4, l = n & 15;
        const size_t base = ((size_t)b * KT_ + kt) * 32;
        k2[base + l]      = make_float2(cx, cy);
        k2[base + 16 + l] = make_float2(cz, cx * cx + cy * cy + cz * cz);
    }
    for (int i = t; i < KS * PH; i += 128) {
        int k = i / PH, p = i % PH;
        float acc = pb1[p];
        acc = fmaf(os[k * 3 + 0], pw1[0 * PH + p], acc);
        acc = fmaf(os[k * 3 + 1], pw1[1 * PH + p], acc);
        acc = fmaf(os[k * 3 + 2], pw1[2 * PH + p], acc);
        ph[i] = gelu_exact(acc);
    }
    __syncthreads();

    if (t < KS * H_) {
        int k = t / H_, h = t % H_;
        float acc = pb2[h];
        #pragma unroll
        for (int p = 0; p < PH; ++p) acc = fmaf(ph[k * PH + p], pw2[p * H_ + h], acc);
        biasb[(size_t)bn * (KS * H_) + t] = acc;
    }
}

// ---------------------------------------------------------------------------
// Kernel 2: nearest-neighbor argmin via V_WMMA_F32_16X16X4_F32.
// Score(q,k) = |k|^2 - 2 q.k  (same argmin as squared distance).
// Operands are pre-swizzled: each lane does ONE coalesced 8-byte load per
// key tile, feeding the WMMA directly (no selects in the hot loop).
// ---------------------------------------------------------------------------
__global__ void nn_kernel(const float* __restrict__ aswz_, const float* __restrict__ kswz_,
                          int* __restrict__ nn)
{
    const v2f* aswz = (const v2f*)aswz_;
    const v2f* kswz = (const v2f*)kswz_;

    const int blk  = blockIdx.x;          // B * (NQ/16) = 8192 blocks
    const int b    = blk >> 11;           // / 2048
    const int qt   = blk & (N_ - 1);
    const int lane = threadIdx.x;
    const int half = lane >> 4;
    const int l16  = lane & 15;

    const v2f a = aswz[((size_t)b * N_ + qt) * 32 + lane];

    float minv[8];
    int   mini[8];
    #pragma unroll
    for (int v = 0; v < 8; ++v) { minv[v] = 3.4e38f; mini[v] = 0; }

    const v2f* kp = kswz + (size_t)b * KT_ * 32;
    for (int kt = 0; kt < KT_; ++kt) {
        const v2f bv = kp[kt * 32 + lane];
        v8f c = {0.f, 0.f, 0.f, 0.f, 0.f, 0.f, 0.f, 0.f};
        c = __builtin_amdgcn_wmma_f32_16x16x4_f32(false, a, false, bv,
                                                  (short)0, c, false, false);
        const int idx = kt * 16 + l16;   // column (key) index held by this lane
        #pragma unroll
        for (int v = 0; v < 8; ++v) {
            bool lt = c[v] < minv[v];    // strict: keep first occurrence
            minv[v] = lt ? c[v] : minv[v];
            mini[v] = lt ? idx  : mini[v];
        }
    }

    // cross-lane argmin within each 16-lane half (tie-break: lower index)
    #pragma unroll
    for (int m = 1; m < 16; m <<= 1) {
        #pragma unroll
        for (int v = 0; v < 8; ++v) {
            float ov = __shfl_xor(minv[v], m, 32);
            int   oi = __shfl_xor(mini[v], m, 32);
            bool take = (ov < minv[v]) || (ov == minv[v] && oi < mini[v]);
            minv[v] = take ? ov : minv[v];
            mini[v] = take ? oi : mini[v];
        }
    }

    if (l16 == 0) {
        const int base = qt * 16 + half * 8;   // rows v (+8 for upper half)
        #pragma unroll
        for (int v = 0; v < 8; ++v)
            nn[(size_t)b * NQ + base + v] = mini[v];
    }
}

// ---------------------------------------------------------------------------
// Kernel 3/5: f32 GEMM  Out[8192,Nd] = A[8192,Kd] @ W[Kd,Nd] + bias, via
// V_WMMA_F32_16X16X4_F32. One wave per 16x16 output tile.
// ---------------------------------------------------------------------------
__global__ void gemm_wmma_kernel(const float* __restrict__ A, const float* __restrict__ W,
                                 const float* __restrict__ bias, float* __restrict__ Out,
                                 int Kd, int Nd)
{
    const int ntiles = Nd >> 4;
    const int mt   = blockIdx.x / ntiles;
    const int nt   = blockIdx.x % ntiles;
    const int lane = threadIdx.x;
    const int half = lane >> 4;
    const int l16  = lane & 15;

    const int m = mt * 16 + l16;
    const int n = nt * 16 + l16;
    const float* arow = A + (size_t)m * Kd;

    v8f c = {0.f, 0.f, 0.f, 0.f, 0.f, 0.f, 0.f, 0.f};
    #pragma unroll 4
    for (int kk = 0; kk < Kd; kk += 4) {
        const int k0 = kk + half * 2;
        v2f a;
        a.x = arow[k0];
        a.y = arow[k0 + 1];
        v2f bv;
        bv.x = W[(size_t)k0 * Nd + n];
        bv.y = W[(size_t)(k0 + 1) * Nd + n];
        c = __builtin_amdgcn_wmma_f32_16x16x4_f32(false, a, false, bv,
                                                  (short)0, c, false, false);
    }

    const float bb = bias[n];
    #pragma unroll
    for (int v = 0; v < 8; ++v) {
        const int orow = mt * 16 + v + half * 8;
        Out[(size_t)orow * Nd + n] = c[v] + bb;
    }
}

// ---------------------------------------------------------------------------
// Kernel 4: gather k/v at nn indices, K=16 attention with pos bias + softmax.
// One thread per (b,h,n).
// ---------------------------------------------------------------------------
__global__ void attn_kernel(const float* __restrict__ qkv, const int* __restrict__ nn,
                            const float* __restrict__ biasb, float* __restrict__ out)
{
    const int g = blockIdx.x * blockDim.x + threadIdx.x;
    if (g >= B_ * H_ * N_) return;
    const int b = g / (H_ * N_);
    const int r = g % (H_ * N_);
    const int h = r / N_;
    const int n = r % N_;

    const float4* qp = (const float4*)(qkv + ((size_t)(b * N_ + n)) * C3 + h * D_);
    float4 q[D_ / 4];
    #pragma unroll
    for (int d = 0; d < D_ / 4; ++d) q[d] = qp[d];

    float s[KS];
    int   idxs[KS];
    const float scale = 0.17677669529663689f;   // 32^-0.5
    for (int k = 0; k < KS; ++k) {
        const int idx = nn[((size_t)b * N_ + n) * KS + k];
        idxs[k] = idx;
        const float4* kp = (const float4*)(qkv + ((size_t)(b * N_ + idx)) * C3 + C_ + h * D_);
        float acc = 0.f;
        #pragma unroll
        for (int d = 0; d < D_ / 4; ++d) {
            float4 kv = kp[d];
            acc = fmaf(q[d].x, kv.x, acc);
            acc = fmaf(q[d].y, kv.y, acc);
            acc = fmaf(q[d].z, kv.z, acc);
            acc = fmaf(q[d].w, kv.w, acc);
        }
        s[k] = acc * scale + biasb[(((size_t)b * N_ + n) * KS + k) * H_ + h];
    }

    float mx = s[0];
    #pragma unroll
    for (int k = 1; k < KS; ++k) mx = fmaxf(mx, s[k]);
    float sum = 0.f;
    #pragma unroll
    for (int k = 0; k < KS; ++k) { s[k] = expf(s[k] - mx); sum += s[k]; }
    const float inv = 1.0f / sum;

    float4 o[D_ / 4];
    #pragma unroll
    for (int d = 0; d < D_ / 4; ++d) o[d] = make_float4(0.f, 0.f, 0.f, 0.f);
    for (int k = 0; k < KS; ++k) {
        const float4* vp = (const float4*)(qkv + ((size_t)(b * N_ + idxs[k])) * C3 + 2 * C_ + h * D_);
        const float p = s[k] * inv;
        #pragma unroll
        for (int d = 0; d < D_ / 4; ++d) {
            float4 vv = vp[d];
            o[d].x = fmaf(p, vv.x, o[d].x);
            o[d].y = fmaf(p, vv.y, o[d].y);
            o[d].z = fmaf(p, vv.z, o[d].z);
            o[d].w = fmaf(p, vv.w, o[d].w);
        }
    }

    float4* op = (float4*)(out + ((size_t)(b * N_ + n)) * C_ + h * D_);
    #pragma unroll
    for (int d = 0; d < D_ / 4; ++d) op[d] = o[d];
}

// ---------------------------------------------------------------------------
extern "C" void kernel_launch(void* const* d_in, const int* in_sizes, int n_in,
                              void* d_out, int out_size, void* d_ws, size_t ws_size,
                              hipStream_t stream)
{
    (void)in_sizes; (void)n_in; (void)out_size; (void)ws_size;

    const float* coords = (const float*)d_in[0];
    const float* x      = (const float*)d_in[1];
    const float* qkv_w  = (const float*)d_in[2];
    const float* qkv_b  = (const float*)d_in[3];
    const float* proj_w = (const float*)d_in[4];
    const float* proj_b = (const float*)d_in[5];
    const float* off_w1 = (const float*)d_in[6];
    const float* off_b1 = (const float*)d_in[7];
    const float* off_w2 = (const float*)d_in[8];
    const float* off_b2 = (const float*)d_in[9];
    const float* pos_w1 = (const float*)d_in[10];
    const float* pos_b1 = (const float*)d_in[11];
    const float* pos_w2 = (const float*)d_in[12];
    const float* pos_b2 = (const float*)d_in[13];

    // workspace layout (floats)
    float* ws    = (float*)d_ws;
    float* qkv   = ws;                                   // 8192*576
    float* aout  = qkv   + (size_t)8192 * C3;            // 8192*192
    float* aswz  = aout  + (size_t)8192 * C_;            // B*N*32*2 (query tiles)
    float* kswz  = aswz  + (size_t)B_ * N_ * 64;         // B*128*32*2 (key tiles)
    float* biasb = kswz  + (size_t)B_ * KT_ * 64;        // 8192*96
    int*   nn    = (int*)(biasb + (size_t)8192 * 96);    // B*NQ ints

    // 1) offset MLP + swizzled WMMA operand packs + pos bias
    prep_kernel<<<B_ * N_, 128, 0, stream>>>(coords, x,
                                             off_w1, off_b1, off_w2, off_b2,
                                             pos_w1, pos_b1, pos_w2, pos_b2,
                                             aswz, kswz, biasb);

    // 2) qkv GEMM: [8192,192] @ [192,576] + b  (WMMA f32 16x16x4)
    gemm_wmma_kernel<<<(8192 / 16) * (C3 / 16), 32, 0, stream>>>(
        x, qkv_w, qkv_b, qkv, C_, C3);

    // 3) nearest-neighbor argmin (WMMA-based score tiles)
    nn_kernel<<<B_ * (NQ / 16), 32, 0, stream>>>(aswz, kswz, nn);

    // 4) gather + K=16 attention with positional bias
    attn_kernel<<<(B_ * H_ * N_) / 64, 64, 0, stream>>>(qkv, nn, biasb, aout);

    // 5) output projection: [8192,192] @ [192,192] + b -> d_out
    gemm_wmma_kernel<<<(8192 / 16) * (C_ / 16), 32, 0, stream>>>(
        aout, proj_w, proj_b, (float*)d_out, C_, C_);
}